// ampPredictor_29626684408068
// MI455X (gfx1250) — compile-verified
//
#include <hip/hip_runtime.h>

typedef float v2f __attribute__((ext_vector_type(2)));
typedef float v4f __attribute__((ext_vector_type(4)));
typedef float v8f __attribute__((ext_vector_type(8)));

static constexpr int kB  = 64;
static constexpr int kL  = 1024;
static constexpr int kD  = 1280;
static constexpr int kH1 = 640;
static constexpr int kH2 = 320;

// Phase 1: per-(batch, L-chunk) partial masked sums over positions [1, len-1).
// 320 threads/block, each owning 4 consecutive floats of the 1280-wide row.
// Non-temporal B128 loads: rep is 335 MB streamed once (> 192 MB L2).
__global__ void pool_partial_kernel(const float* __restrict__ rep,
                                    const int* __restrict__ blen,
                                    float* __restrict__ part,
                                    int chunk_len) {
  const int b = blockIdx.x;
  const int c = blockIdx.y;
  const int d = threadIdx.x * 4;
  const int len = blen[b];

  int lo = c * chunk_len;
  int hi = lo + chunk_len;
  if (lo < 1) lo = 1;
  const int ub = len - 1;           // exclusive upper bound
  if (hi > ub) hi = ub;

  v4f acc = {0.f, 0.f, 0.f, 0.f};
  const v4f* row = (const v4f*)(rep + (size_t)b * kL * kD + d);
  for (int l = lo; l < hi; ++l) {
    acc += __builtin_nontemporal_load(row + (size_t)l * (kD / 4));
  }
  // Unconditional write (zero if empty intersection) -> no memset needed.
  *(v4f*)(part + ((size_t)c * kB + b) * kD + d) = acc;
}

// Phase 2: deterministic fixed-order reduction over chunks + divide by count.
__global__ void pool_reduce_kernel(const float* __restrict__ part,
                                   const int* __restrict__ blen,
                                   float* __restrict__ pooled,
                                   int chunks) {
  const int b = blockIdx.x;
  const int d = threadIdx.x * 4;
  v4f acc = {0.f, 0.f, 0.f, 0.f};
  for (int c = 0; c < chunks; ++c) {
    acc += *(const v4f*)(part + ((size_t)c * kB + b) * kD + d);
  }
  int cnt = blen[b] - 2;            // len >= 3 guaranteed; clamp anyway
  if (cnt < 1) cnt = 1;
  acc *= (1.0f / (float)cnt);
  *(v4f*)(pooled + (size_t)b * kD + d) = acc;
}

// Out[m0:m0+16, n0:n0+16] = relu(A[m0:+16, :K] @ W[:K, n0:+16] + bias)
// One wave (32 lanes) per 16x16 tile; K-loop of V_WMMA_F32_16X16X4_F32.
__global__ void wmma_gemm_relu_kernel(const float* __restrict__ A,
                                      const float* __restrict__ W,   // K x N row-major
                                      const float* __restrict__ bias,
                                      float* __restrict__ Out,
                                      int K, int N) {
  const int n0   = blockIdx.x * 16;
  const int m0   = blockIdx.y * 16;
  const int lane = threadIdx.x;       // 0..31, full wave, no divergence
  const int half = lane >> 4;
  const int ln   = lane & 15;

  v8f acc = {0.f, 0.f, 0.f, 0.f, 0.f, 0.f, 0.f, 0.f};
  const float* arow = A + (size_t)(m0 + ln) * K;   // A fragment row for this lane

  for (int k0 = 0; k0 < K; k0 += 4) {
    const int ka = k0 + 2 * half;                  // even -> 8B-aligned v2f load
    v2f afrag = *(const v2f*)(arow + ka);          // A[ln][ka], A[ln][ka+1]
    v2f bfrag;
    bfrag.x = W[(size_t)ka * N + n0 + ln];         // B[ka][ln]
    bfrag.y = W[(size_t)(ka + 1) * N + n0 + ln];   // B[ka+1][ln]
    acc = __builtin_amdgcn_wmma_f32_16x16x4_f32(
        /*neg_a=*/false, afrag, /*neg_b=*/false, bfrag,
        /*c_mod=*/(short)0, acc, /*reuse_a=*/false, /*reuse_b=*/false);
  }

  const float bv = bias[n0 + ln];
#pragma unroll
  for (int r = 0; r < 8; ++r) {
    float v = acc[r] + bv;
    v = v > 0.f ? v : 0.f;                         // ReLU
    Out[(size_t)(m0 + r + 8 * half) * N + n0 + ln] = v;   // C/D layout: row r+8*half, col ln
  }
}

// Final 320 -> 2 layer: 128 scalar dot products (negligible work).
__global__ void final_fc_kernel(const float* __restrict__ emb,
                                const float* __restrict__ W3,  // 320 x 2
                                const float* __restrict__ b3,
                                float* __restrict__ y) {
  const int tid = threadIdx.x;       // 0..127
  const int b = tid >> 1, j = tid & 1;
  float s = b3[j];
  const float* e = emb + (size_t)b * kH2;
  for (int k = 0; k < kH2; ++k) s += e[k] * W3[k * 2 + j];
  y[tid] = s;
}

extern "C" void kernel_launch(void* const* d_in, const int* in_sizes, int n_in,
                              void* d_out, int out_size, void* d_ws, size_t ws_size,
                              hipStream_t stream) {
  const float* rep  = (const float*)d_in[0];
  const int*   blen = (const int*)d_in[1];
  const float* W1   = (const float*)d_in[2];
  const float* b1   = (const float*)d_in[3];
  const float* W2   = (const float*)d_in[4];
  const float* b2   = (const float*)d_in[5];
  const float* W3   = (const float*)d_in[6];
  const float* b3   = (const float*)d_in[7];
  float* out = (float*)d_out;
  float* ws  = (float*)d_ws;

  // Pick L-chunk count for parallelism (16 -> 1024 blocks streaming rep),
  // backing off only if the workspace is unexpectedly small. Deterministic.
  int chunks = 16;
  auto need = [](int c) -> size_t {
    return ((size_t)c * kB * kD + (size_t)kB * kD + (size_t)kB * kH1) * sizeof(float);
  };
  while (chunks > 1 && need(chunks) > ws_size) chunks >>= 1;
  const int chunk_len = kL / chunks;

  float* part   = ws;                                  // [chunks][B][D]
  float* pooled = part + (size_t)chunks * kB * kD;     // [B][D]
  float* h      = pooled + (size_t)kB * kD;            // [B][H1]
  float* y      = out;                                 // [B][2]   (output 0)
  float* emb    = out + (size_t)kB * 2;                // [B][H2]  (output 1)

  pool_partial_kernel<<<dim3(kB, chunks), kD / 4, 0, stream>>>(rep, blen, part, chunk_len);
  pool_reduce_kernel<<<kB, kD / 4, 0, stream>>>(part, blen, pooled, chunks);
  // h = relu(pooled @ W1 + b1)   : M=64, K=1280, N=640 -> 4x40 tiles
  wmma_gemm_relu_kernel<<<dim3(kH1 / 16, kB / 16), 32, 0, stream>>>(pooled, W1, b1, h, kD, kH1);
  // emb = relu(h @ W2 + b2)      : M=64, K=640, N=320 -> 4x20 tiles
  wmma_gemm_relu_kernel<<<dim3(kH2 / 16, kB / 16), 32, 0, stream>>>(h, W2, b2, emb, kH1, kH2);
  // y = emb @ W3 + b3
  final_fc_kernel<<<1, 128, 0, stream>>>(emb, W3, b3, y);
}